// InstanceSegmentationLoss_67362267070604
// MI455X (gfx1250) — compile-verified
//
#include <hip/hip_runtime.h>

#define KIDS 64
#define NBINS (KIDS * KIDS)

typedef float v2f __attribute__((ext_vector_type(2)));
typedef float v8f __attribute__((ext_vector_type(8)));

// ---------------------------------------------------------------------------
// Kernel 0: zero the u64 sum-of-squares accumulator (everything else in ws is
// fully overwritten each call, so no other zeroing needed).
// ---------------------------------------------------------------------------
__global__ __launch_bounds__(32) void zero_sq_kernel(unsigned long long* gsq) {
    if (threadIdx.x == 0) *gsq = 0ull;
}

// ---------------------------------------------------------------------------
// Kernel 1: streaming pass. float4 loads, LDS-private 64x64 histogram,
// exact integer sum of squared differences.
// ---------------------------------------------------------------------------
__global__ __launch_bounds__(256) void hist_partial_kernel(
    const float4* __restrict__ pred, const float4* __restrict__ tru,
    unsigned long long* __restrict__ gsq, unsigned int* __restrict__ partial,
    int n4) {
    __shared__ unsigned int lh[NBINS];          // 16 KB private histogram
    __shared__ unsigned long long ssq;
    const int tid = threadIdx.x;

    for (int i = tid; i < NBINS; i += 256) lh[i] = 0u;
    if (tid == 0) ssq = 0ull;
    __syncthreads();

    unsigned int lsq = 0u;                      // <= 8192 px * 3969 fits u32
    int idx = blockIdx.x * 256 + tid;
    const int stride = gridDim.x * 256;
    for (; idx < n4; idx += stride) {
        float4 p = pred[idx];
        float4 t = tru[idx];
        int p0 = (int)p.x, p1 = (int)p.y, p2 = (int)p.z, p3 = (int)p.w;
        int t0 = (int)t.x, t1 = (int)t.y, t2 = (int)t.z, t3 = (int)t.w;
        int d0 = p0 - t0, d1 = p1 - t1, d2 = p2 - t2, d3 = p3 - t3;
        lsq += (unsigned int)(d0 * d0 + d1 * d1 + d2 * d2 + d3 * d3);
        atomicAdd(&lh[p0 * KIDS + t0], 1u);
        atomicAdd(&lh[p1 * KIDS + t1], 1u);
        atomicAdd(&lh[p2 * KIDS + t2], 1u);
        atomicAdd(&lh[p3 * KIDS + t3], 1u);
    }

    atomicAdd(&ssq, (unsigned long long)lsq);   // DS 64-bit atomic
    __syncthreads();
    if (tid == 0) atomicAdd(gsq, ssq);          // one global u64 atomic/block

    // Non-atomic flush of the private histogram to this block's partial slot.
    unsigned int* mypart = partial + (size_t)blockIdx.x * NBINS;
    for (int i = tid; i < NBINS; i += 256) mypart[i] = lh[i];
}

// ---------------------------------------------------------------------------
// Kernel 2: sum the per-block partial histograms (coalesced, no atomics).
// ---------------------------------------------------------------------------
__global__ __launch_bounds__(256) void hist_reduce_kernel(
    const unsigned int* __restrict__ partial, unsigned int* __restrict__ ghist,
    int nblk) {
    int bin = blockIdx.x * 256 + threadIdx.x;
    if (bin < NBINS) {
        unsigned int acc = 0u;
        for (int p = 0; p < nblk; ++p)
            acc += partial[(size_t)p * NBINS + bin];
        ghist[bin] = acc;
    }
}

// ---------------------------------------------------------------------------
// Kernel 3: one wave32. area_p / area_t via V_WMMA_F32_16X16X4_F32
// (A = counts tile per documented 16x4 f32 layout, B = ones), then the
// 64x64 IoU maxima and the final scalar loss.
// ---------------------------------------------------------------------------
__global__ __launch_bounds__(32) void finalize_kernel(
    const unsigned int* __restrict__ ghist,
    const unsigned long long* __restrict__ gsq,
    float* __restrict__ out, float inv_n) {
    __shared__ float sc[NBINS];
    __shared__ float s_ap[KIDS], s_at[KIDS];
    __shared__ int   s_pp[KIDS], s_pt[KIDS];
    __shared__ float s_maxt[KIDS], s_maxp[KIDS];

    const int lane = threadIdx.x;
    for (int i = lane; i < NBINS; i += 32) sc[i] = (float)ghist[i];
    __syncthreads();

    // A-matrix 16x4 f32 layout: lanes 0-15 -> M=lane, VGPR0/1 = K 0/1;
    // lanes 16-31 -> M=lane-16, VGPR0/1 = K 2/3. B = ones (layout-invariant).
    v2f onesb; onesb.x = 1.0f; onesb.y = 1.0f;
    const int mloc = lane & 15;
    const int kloc = (lane < 16) ? 0 : 2;

    // area_p[r] = sum_t counts[r][t]  (row sums)
    for (int mt = 0; mt < 4; ++mt) {
        int row = mt * 16 + mloc;
        v8f acc = {};
        for (int k0 = 0; k0 < KIDS; k0 += 4) {
            v2f a;
            a.x = sc[row * KIDS + k0 + kloc];
            a.y = sc[row * KIDS + k0 + kloc + 1];
            acc = __builtin_amdgcn_wmma_f32_16x16x4_f32(
                false, a, false, onesb, (short)0, acc, false, false);
        }
        // C/D layout: VGPR v -> M=v (lanes 0-15), M=v+8 (lanes 16-31)
        if (lane == 0)
            for (int v = 0; v < 8; ++v) s_ap[mt * 16 + v] = acc[v];
        if (lane == 16)
            for (int v = 0; v < 8; ++v) s_ap[mt * 16 + 8 + v] = acc[v];
    }

    // area_t[c] = sum_p counts[p][c]  (row sums of the transpose)
    for (int mt = 0; mt < 4; ++mt) {
        int col = mt * 16 + mloc;
        v8f acc = {};
        for (int k0 = 0; k0 < KIDS; k0 += 4) {
            v2f a;
            a.x = sc[(k0 + kloc) * KIDS + col];
            a.y = sc[(k0 + kloc + 1) * KIDS + col];
            acc = __builtin_amdgcn_wmma_f32_16x16x4_f32(
                false, a, false, onesb, (short)0, acc, false, false);
        }
        if (lane == 0)
            for (int v = 0; v < 8; ++v) s_at[mt * 16 + v] = acc[v];
        if (lane == 16)
            for (int v = 0; v < 8; ++v) s_at[mt * 16 + 8 + v] = acc[v];
    }
    __syncthreads();

    for (int i = lane; i < KIDS; i += 32) {
        s_pp[i] = (s_ap[i] > 0.0f) && (i != 0);
        s_pt[i] = (s_at[i] > 0.0f) && (i != 0);
    }
    __syncthreads();

    // Each lane owns rows/cols {lane, lane+32}.
    for (int rr = 0; rr < 2; ++rr) {
        int r = lane + rr * 32;
        float ap = s_ap[r];
        float m = 0.0f;
        for (int t = 0; t < KIDS; ++t) {
            if (!s_pt[t]) continue;
            float inter = sc[r * KIDS + t];
            float uni = ap + s_at[t] - inter;
            float iou = (uni > 0.0f) ? (inter / uni) : 0.0f;
            m = fmaxf(m, iou);
        }
        s_maxt[r] = m;

        int c = r;
        float at = s_at[c];
        float mm = 0.0f;
        for (int p = 0; p < KIDS; ++p) {
            if (!s_pp[p]) continue;
            float inter = sc[p * KIDS + c];
            float uni = s_ap[p] + at - inter;
            float iou = (uni > 0.0f) ? (inter / uni) : 0.0f;
            mm = fmaxf(mm, iou);
        }
        s_maxp[c] = mm;
    }
    __syncthreads();

    if (lane == 0) {
        float lp = 0.0f, lt = 0.0f;
        int ni = 0;
        for (int i = 0; i < KIDS; ++i) {
            if (s_pp[i]) { lp += 1.0f - s_maxt[i]; ++ni; }
            if (s_pt[i]) { lt += 1.0f - s_maxp[i]; ++ni; }
        }
        double mse = (double)(*gsq) * (double)inv_n / 1000.0;
        float total = (float)mse + lp + lt;
        out[0] = (ni > 0) ? total : 0.0f;
    }
}

// ---------------------------------------------------------------------------
// Host launcher. ws layout: [0..15] u64 sq (padded) | 16KB ghist | partials.
// ---------------------------------------------------------------------------
extern "C" void kernel_launch(void* const* d_in, const int* in_sizes, int n_in,
                              void* d_out, int out_size, void* d_ws, size_t ws_size,
                              hipStream_t stream) {
    const float* pred = (const float*)d_in[0];
    const float* tru  = (const float*)d_in[1];
    float* out = (float*)d_out;

    unsigned long long* gsq = (unsigned long long*)d_ws;
    unsigned int* ghist   = (unsigned int*)((char*)d_ws + 16);
    unsigned int* partial = ghist + NBINS;

    const int n  = in_sizes[0];     // 4096*4096 = 16777216
    const int n4 = n >> 2;

    // Adapt the number of streaming blocks to workspace size (cap 512).
    const size_t fixed = 16 + (size_t)NBINS * sizeof(unsigned int);
    int nblk = 1;
    if (ws_size > fixed) {
        size_t avail = (ws_size - fixed) / ((size_t)NBINS * sizeof(unsigned int));
        nblk = (int)(avail < 512 ? avail : 512);
        if (nblk < 1) nblk = 1;
    }

    zero_sq_kernel<<<1, 32, 0, stream>>>(gsq);
    hist_partial_kernel<<<nblk, 256, 0, stream>>>(
        (const float4*)pred, (const float4*)tru, gsq, partial, n4);
    hist_reduce_kernel<<<(NBINS + 255) / 256, 256, 0, stream>>>(partial, ghist, nblk);
    finalize_kernel<<<1, 32, 0, stream>>>(ghist, gsq, out, 1.0f / (float)n);
}